// BipartiteGcn_37014028157513
// MI455X (gfx1250) — compile-verified
//
#include <hip/hip_runtime.h>
#include <stdint.h>

namespace {
constexpr int kNE = 100000;   // entities
constexpr int kNR = 1000;     // relations
constexpr int kD  = 64;       // embedding width
constexpr int kE  = 1000000;  // edges
constexpr int kHalfR = 500;   // relation rows per LDS pass (500*64*4 = 128 KB)
constexpr int kNED = kNE * kD;
constexpr int kNRD = kNR * kD;
}

// ---- gfx1250 async global->LDS copy (ASYNCcnt path) -----------------------
typedef int v4i __attribute__((vector_size(16)));
typedef __attribute__((address_space(1))) v4i* gv4_t;
typedef __attribute__((address_space(3))) v4i* lv4_t;

__device__ __forceinline__ void async_cp16_g2l(void* lds_dst, const void* gsrc) {
#if __has_builtin(__builtin_amdgcn_global_load_async_to_lds_b128)
  __builtin_amdgcn_global_load_async_to_lds_b128(
      (gv4_t)(uintptr_t)gsrc,
      (lv4_t)(uint32_t)(uintptr_t)lds_dst,
      /*offset=*/0, /*cpol=*/0);
#else
  uint32_t loff = (uint32_t)(uintptr_t)lds_dst;
  asm volatile("global_load_async_to_lds_b128 %0, %1, off"
               :: "v"(loff), "v"(gsrc) : "memory");
#endif
}

__device__ __forceinline__ void wait_async0() {
#if __has_builtin(__builtin_amdgcn_s_wait_asynccnt)
  __builtin_amdgcn_s_wait_asynccnt(0);
#else
  asm volatile("s_wait_asynccnt 0" ::: "memory");
#endif
}

// ---- zero output + degree counters ----------------------------------------
__global__ void k_zero(float* __restrict__ out, unsigned n_out,
                       unsigned* __restrict__ deg, unsigned n_deg) {
  unsigned i = blockIdx.x * blockDim.x + threadIdx.x;
  const unsigned stride = gridDim.x * blockDim.x;
  const unsigned total = n_out + n_deg;
  for (; i < total; i += stride) {
    if (i < n_out) out[i] = 0.0f;
    else           deg[i - n_out] = 0u;
  }
}

// ---- degree histograms (relation side privatized in LDS) ------------------
__global__ void k_hist(const int* __restrict__ e_send, const int* __restrict__ e_recv,
                       const int* __restrict__ r_send, const int* __restrict__ r_recv,
                       unsigned* __restrict__ deg_er, unsigned* __restrict__ deg_es,
                       unsigned* __restrict__ deg_rr, unsigned* __restrict__ deg_rs) {
  __shared__ unsigned h[2 * kNR];  // 8 KB
  for (int i = threadIdx.x; i < 2 * kNR; i += blockDim.x) h[i] = 0u;
  __syncthreads();
  unsigned i = blockIdx.x * blockDim.x + threadIdx.x;
  const unsigned stride = gridDim.x * blockDim.x;
  for (; i < (unsigned)kE; i += stride) {
    atomicAdd(&deg_er[e_recv[i]], 1u);   // deg over e_receiver
    atomicAdd(&deg_es[e_send[i]], 1u);   // deg over e_sender
    atomicAdd(&h[r_recv[i]], 1u);        // deg over r_receiver
    atomicAdd(&h[kNR + r_send[i]], 1u);  // deg over r_sender
  }
  __syncthreads();
  for (int j = threadIdx.x; j < kNR; j += blockDim.x) {
    unsigned a = h[j], b = h[kNR + j];
    if (a) atomicAdd(&deg_rr[j], a);
    if (b) atomicAdd(&deg_rs[j], b);
  }
}

// ---- 1 / max(deg, 1) -------------------------------------------------------
__global__ void k_inv(const unsigned* __restrict__ deg, float* __restrict__ inv, unsigned n) {
  unsigned i = blockIdx.x * blockDim.x + threadIdx.x;
  const unsigned stride = gridDim.x * blockDim.x;
  for (; i < n; i += stride) {
    unsigned d = deg[i];
    inv[i] = 1.0f / (float)(d ? d : 1u);
  }
}

// ---- build relu(X + bias) tables ------------------------------------------
__global__ void k_tables(const float* __restrict__ Ef, const float* __restrict__ Eb,
                         const float* __restrict__ Rf, const float* __restrict__ Rb,
                         const float* __restrict__ bEf, const float* __restrict__ bEb,
                         const float* __restrict__ bRf, const float* __restrict__ bRb,
                         float* __restrict__ Sf, float* __restrict__ Sb,
                         float* __restrict__ Tf, float* __restrict__ Tb) {
  unsigned i = blockIdx.x * blockDim.x + threadIdx.x;
  const unsigned stride = gridDim.x * blockDim.x;
  const unsigned total = 2u * kNED + 2u * kNRD;
  for (; i < total; i += stride) {
    if (i < (unsigned)kNED) {
      Sf[i] = fmaxf(Ef[i] + bEf[i & 63u], 0.0f);
    } else if (i < 2u * kNED) {
      unsigned j = i - kNED;
      Sb[j] = fmaxf(Eb[j] + bEb[j & 63u], 0.0f);
    } else if (i < 2u * kNED + (unsigned)kNRD) {
      unsigned j = i - 2u * kNED;
      Tf[j] = fmaxf(Rf[j] + bRf[j & 63u], 0.0f);
    } else {
      unsigned j = i - 2u * kNED - kNRD;
      Tb[j] = fmaxf(Rb[j] + bRb[j & 63u], 0.0f);
    }
  }
}

// ---- entity-output scatter: gather relation rows from LDS-staged table ----
// stream 0: out[e_recv] += Tf[r_send] * inv_er[e_recv]
// stream 1: out[e_send] += Tb[r_recv] * inv_es[e_send]
__global__ void __launch_bounds__(256) k_scatter_entity(
    const float* __restrict__ Tf, const float* __restrict__ Tb,
    const int* __restrict__ e_send, const int* __restrict__ e_recv,
    const int* __restrict__ r_send, const int* __restrict__ r_recv,
    const float* __restrict__ inv_er, const float* __restrict__ inv_es,
    float* __restrict__ out, int nb_per_stream, int chunk) {
  __shared__ float s_tab[kHalfR * kD];  // 128 KB
  const int strm = (blockIdx.x >= (unsigned)nb_per_stream) ? 1 : 0;
  const int b = (int)blockIdx.x - strm * nb_per_stream;
  const float* tab  = strm ? Tb     : Tf;
  const int*   ridx = strm ? r_recv : r_send;
  const int*   vidx = strm ? e_send : e_recv;
  const float* inv  = strm ? inv_es : inv_er;
  const int e0 = b * chunk;
  const int e1 = min(kE, e0 + chunk);
  const int lane = threadIdx.x & 31;
  const int wid  = threadIdx.x >> 5;
  const int nw   = blockDim.x >> 5;
  for (int half = 0; half < 2; ++half) {
    const int rbase = half * kHalfR;
    // async-stage 500 table rows (128 KB) into LDS
    const float* src = tab + (size_t)rbase * kD;
    for (int i = (int)threadIdx.x * 4; i < kHalfR * kD; i += (int)blockDim.x * 4)
      async_cp16_g2l(&s_tab[i], &src[i]);
    wait_async0();
    __syncthreads();
    for (int e = e0 + wid; e < e1; e += nw) {
      const unsigned rl = (unsigned)(ridx[e] - rbase);
      if (rl < (unsigned)kHalfR) {
        const int v = vidx[e];
        const float sc = inv[v];
        const float2 t = *(const float2*)&s_tab[rl * kD + lane * 2];
        float* o = out + (size_t)v * kD + lane * 2;
        atomicAdd(o,     t.x * sc);
        atomicAdd(o + 1, t.y * sc);
      }
    }
    __syncthreads();  // all reads of s_tab done before restaging
  }
}

// ---- relation-output scatter: LDS-privatized accumulators -----------------
// out[NE + r_recv] += Sf[e_send] * inv_rr[r_recv]
// out[NE + r_send] += Sb[e_recv] * inv_rs[r_send]
__global__ void __launch_bounds__(256) k_scatter_rel(
    const float* __restrict__ Sf, const float* __restrict__ Sb,
    const int* __restrict__ e_send, const int* __restrict__ e_recv,
    const int* __restrict__ r_send, const int* __restrict__ r_recv,
    const float* __restrict__ inv_rr, const float* __restrict__ inv_rs,
    float* __restrict__ out, int chunk) {
  __shared__ float s_acc[kHalfR * kD];  // 128 KB
  const int e0 = (int)blockIdx.x * chunk;
  const int e1 = min(kE, e0 + chunk);
  const int lane = threadIdx.x & 31;
  const int wid  = threadIdx.x >> 5;
  const int nw   = blockDim.x >> 5;
  for (int half = 0; half < 2; ++half) {
    const int rbase = half * kHalfR;
    for (int i = threadIdx.x; i < kHalfR * kD; i += blockDim.x) s_acc[i] = 0.0f;
    __syncthreads();
    for (int e = e0 + wid; e < e1; e += nw) {
      {
        const int r = r_recv[e];
        const unsigned rl = (unsigned)(r - rbase);
        if (rl < (unsigned)kHalfR) {
          const int a = e_send[e];
          const float sc = inv_rr[r];
          const float2 x = *(const float2*)&Sf[(size_t)a * kD + lane * 2];
          atomicAdd(&s_acc[rl * kD + lane * 2],     x.x * sc);
          atomicAdd(&s_acc[rl * kD + lane * 2 + 1], x.y * sc);
        }
      }
      {
        const int r = r_send[e];
        const unsigned rl = (unsigned)(r - rbase);
        if (rl < (unsigned)kHalfR) {
          const int a = e_recv[e];
          const float sc = inv_rs[r];
          const float2 x = *(const float2*)&Sb[(size_t)a * kD + lane * 2];
          atomicAdd(&s_acc[rl * kD + lane * 2],     x.x * sc);
          atomicAdd(&s_acc[rl * kD + lane * 2 + 1], x.y * sc);
        }
      }
    }
    __syncthreads();
    float* obase = out + (size_t)kNE * kD + (size_t)rbase * kD;
    for (int i = threadIdx.x; i < kHalfR * kD; i += blockDim.x) {
      const float v = s_acc[i];
      if (v != 0.0f) atomicAdd(&obase[i], v);
    }
    __syncthreads();
  }
}

extern "C" void kernel_launch(void* const* d_in, const int* in_sizes, int n_in,
                              void* d_out, int out_size, void* d_ws, size_t ws_size,
                              hipStream_t stream) {
  (void)in_sizes; (void)n_in; (void)ws_size;
  const float* Ef  = (const float*)d_in[0];
  const float* Eb  = (const float*)d_in[1];
  const float* Rf  = (const float*)d_in[2];
  const float* Rb  = (const float*)d_in[3];
  const float* bEf = (const float*)d_in[4];
  const float* bEb = (const float*)d_in[5];
  const float* bRf = (const float*)d_in[6];
  const float* bRb = (const float*)d_in[7];
  const int* e_send = (const int*)d_in[8];
  const int* e_recv = (const int*)d_in[9];
  const int* r_send = (const int*)d_in[10];
  const int* r_recv = (const int*)d_in[11];
  float* out = (float*)d_out;

  // workspace layout (all 16B aligned): Sf | Sb | Tf | Tb | deg[4] | inv[4]
  float* Sf = (float*)d_ws;
  float* Sb = Sf + kNED;
  float* Tf = Sb + kNED;
  float* Tb = Tf + kNRD;
  unsigned* deg = (unsigned*)(Tb + kNRD);
  const unsigned n_deg = 2u * kNE + 2u * kNR;
  float* inv = (float*)(deg + n_deg);
  unsigned* deg_er = deg;
  unsigned* deg_es = deg + kNE;
  unsigned* deg_rr = deg + 2 * kNE;
  unsigned* deg_rs = deg + 2 * kNE + kNR;
  float* inv_er = inv;
  float* inv_es = inv + kNE;
  float* inv_rr = inv + 2 * kNE;
  float* inv_rs = inv + 2 * kNE + kNR;

  k_zero<<<2048, 256, 0, stream>>>(out, (unsigned)out_size, deg, n_deg);
  k_hist<<<1024, 256, 0, stream>>>(e_send, e_recv, r_send, r_recv,
                                   deg_er, deg_es, deg_rr, deg_rs);
  k_inv<<<(n_deg + 255u) / 256u, 256, 0, stream>>>(deg, inv, n_deg);
  k_tables<<<4096, 256, 0, stream>>>(Ef, Eb, Rf, Rb, bEf, bEb, bRf, bRb,
                                     Sf, Sb, Tf, Tb);
  const int nbs = 128;
  const int chunk = (kE + nbs - 1) / nbs;
  k_scatter_entity<<<2 * nbs, 256, 0, stream>>>(Tf, Tb, e_send, e_recv, r_send, r_recv,
                                                inv_er, inv_es, out, nbs, chunk);
  k_scatter_rel<<<nbs, 256, 0, stream>>>(Sf, Sb, e_send, e_recv, r_send, r_recv,
                                         inv_rr, inv_rs, out, chunk);
}